// Augment_37349035606788
// MI455X (gfx1250) — compile-verified
//
#include <hip/hip_runtime.h>
#include <cmath>

// ---------------------------------------------------------------------------
// Fused augmentation for MI455X (gfx1250, wave32).
// Memory-bound (~256MB traffic @ 23.3TB/s => ~11-20us floor): single fused
// pass, per-batch params precomputed in a tiny setup kernel (uniform SMEM loads
// in the hot kernel), per-pixel 3x3 color matvec on the matrix pipe via
// V_WMMA_F32_16X16X4_F32, gamma via raw v_log_f32/v_exp_f32 (domain-safe since
// base is clamped to [0,1] and gamma > 0), half-wave exchanges via
// ds_swizzle_b32 immediates (no index VGPR).
// ---------------------------------------------------------------------------

typedef __attribute__((ext_vector_type(2))) float v2f;
typedef __attribute__((ext_vector_type(8))) float v8f;

#define DEGF 0.017453292519943295f
#define PSTRIDE 80   // floats per batch in workspace: 16 params + 32 A0 + 32 A1

__device__ __forceinline__ void mm3(const float* A, const float* B, float* C) {
#pragma unroll
  for (int i = 0; i < 3; ++i)
#pragma unroll
    for (int j = 0; j < 3; ++j)
      C[i * 3 + j] = A[i * 3 + 0] * B[0 * 3 + j] +
                     A[i * 3 + 1] * B[1 * 3 + j] +
                     A[i * 3 + 2] * B[2 * 3 + j];
}

// pow(x, g) for x in [0,1], g > 0: two trans-pipe ops + one mul.
// x==0 -> log2=-inf -> g*-inf=-inf -> exp2=0 (matches reference 0^g=0).
__device__ __forceinline__ float pow01(float x, float g) {
#if __has_builtin(__builtin_amdgcn_logf) && __has_builtin(__builtin_amdgcn_exp2f)
  return __builtin_amdgcn_exp2f(g * __builtin_amdgcn_logf(x));  // v_log_f32 is log2
#else
  return exp2f(g * log2f(x));
#endif
}

// lane-swizzle helpers (group-of-32 DS_SWIZZLE patterns, immediate-encoded)
__device__ __forceinline__ float swz_and15(float x) {   // src = lane & 15
  return __int_as_float(__builtin_amdgcn_ds_swizzle(__float_as_int(x), 0x000F));
}
__device__ __forceinline__ float swz_xor16(float x) {   // src = lane ^ 16
  return __int_as_float(__builtin_amdgcn_ds_swizzle(__float_as_int(x), 0x401F));
}

// One thread per batch: build geometry matrix M, cutout/gamma params, and the
// color matrix Mc pre-swizzled into the WMMA 16x4 f32 A-operand lane layout
// (VGPR0: lanes0-15 = A[M][K=0], lanes16-31 = A[M][K=2]; VGPR1: K=1 / K=3).
__global__ void aug_setup(const float* __restrict__ geom,
                          const float* __restrict__ color,
                          const float* __restrict__ cut,
                          float* __restrict__ ws, int B) {
  int b = blockIdx.x * blockDim.x + threadIdx.x;
  if (b >= B) return;
  const float* g  = geom  + b * 8;
  const float* cu = color + b * 4;
  const float* ct = cut   + b * 5;

  // ---- geometry: M = T @ RS @ Rx @ Ry @ F ----
  float flip = (g[0] > 0.5f) ? -1.f : 1.f;
  float tilt = (g[1] * 2.f - 1.f) * (15.f * DEGF);
  float pan  = (g[2] * 2.f - 1.f) * (15.f * DEGF);
  float rot  = (g[3] * 2.f - 1.f) * (15.f * DEGF);
  float sc   = 1.f + (g[4] * 2.f - 1.f) * 0.1f;
  float tx   = (g[5] * 2.f - 1.f) * 0.2f;
  float ty   = (g[6] * 2.f - 1.f) * 0.2f;

  float F[9]  = {flip, 0, 0, 0, 1, 0, 0, 0, 1};
  float ctl = cosf(tilt), stl = sinf(tilt);
  float Rx[9] = {1, 0, 0, 0, ctl, -stl, 0, stl, ctl};
  float cp = cosf(pan), sp = sinf(pan);
  float Ry[9] = {cp, 0, sp, 0, 1, 0, -sp, 0, cp};
  float cr = cosf(rot), sr = sinf(rot);
  float RS[9] = {sc * cr, -sc * sr, 0, sc * sr, sc * cr, 0, 0, 0, 1};
  float T[9]  = {1, 0, tx, 0, 1, ty, 0, 0, 1};
  float t0[9], t1[9], M[9];
  mm3(Ry, F, t0);
  mm3(Rx, t0, t1);
  mm3(RS, t1, t0);
  mm3(T, t0, M);

  float* P = ws + (size_t)b * PSTRIDE;
#pragma unroll
  for (int i = 0; i < 9; ++i) P[i] = M[i];
  P[9]  = 1.f + (cu[3] * 2.f - 1.f) * 0.2f;          // gamma
  P[10] = (ct[0] < 0.5f) ? 1.f : 0.f;                // cutout apply
  P[11] = ct[1];                                     // cx
  P[12] = ct[2];                                     // cy
  P[13] = (0.3f + 0.2f * ct[3]) * 0.5f;              // cw/2
  P[14] = (0.3f + 0.2f * ct[4]) * 0.5f;              // ch/2
  P[15] = 0.f;

  // ---- color: Mc = bright * (Sm @ Rh) ----
  float theta = (cu[0] * 2.f - 1.f) * (10.f * DEGF);
  float c = cosf(theta), s = sinf(theta);
  const float a = 0.5773502691896258f;               // 1/sqrt(3)
  float Kh[9] = {0, -a, a, a, 0, -a, -a, a, 0};
  float Rh[9];
#pragma unroll
  for (int i = 0; i < 3; ++i)
#pragma unroll
    for (int j = 0; j < 3; ++j)
      Rh[i * 3 + j] = c * (i == j ? 1.f : 0.f) + s * Kh[i * 3 + j] + (1.f - c) * (1.f / 3.f);
  float sat = 1.f + (cu[1] * 2.f - 1.f) * 0.4f;
  float lum[3] = {0.299f, 0.587f, 0.114f};
  float Sm[9];
#pragma unroll
  for (int i = 0; i < 3; ++i)
#pragma unroll
    for (int j = 0; j < 3; ++j)
      Sm[i * 3 + j] = sat * (i == j ? 1.f : 0.f) + (1.f - sat) * lum[j];
  float Mc[9];
  mm3(Sm, Rh, Mc);
  float bright = 1.f + (cu[2] * 2.f - 1.f) * 0.1f;
#pragma unroll
  for (int i = 0; i < 9; ++i) Mc[i] *= bright;

  // Pre-swizzle Mc into the WMMA A-operand (16x4 f32) per-lane layout.
  for (int l = 0; l < 32; ++l) {
    float a0 = 0.f, a1 = 0.f;
    if (l < 16) {
      if (l < 3) { a0 = Mc[l * 3 + 0]; a1 = Mc[l * 3 + 1]; }   // K=0 / K=1
    } else {
      int m = l - 16;
      if (m < 3) { a0 = Mc[m * 3 + 2]; }                        // K=2 / K=3(=0)
    }
    P[16 + l] = a0;
    P[48 + l] = a1;
  }
}

__global__ __launch_bounds__(256) void aug_main(
    const float* __restrict__ img, const float* __restrict__ ws,
    float* __restrict__ out, int H, int W, int wshift,
    float invWm1, float invHm1) {
  const int b = blockIdx.y;
  const float* __restrict__ P = ws + (size_t)b * PSTRIDE;   // uniform -> s_load
  const int HW  = H * W;
  int idx = blockIdx.x * blockDim.x + threadIdx.x;
  const int sidx = (idx < HW) ? idx : (HW - 1);   // no early-exit: keep EXEC all-1 for WMMA
  const int iy = (wshift >= 0) ? (sidx >> wshift) : (sidx / W);
  const int ix = sidx - iy * W;
  const int lane = threadIdx.x & 31;

  const float M0 = P[0], M1 = P[1], M2 = P[2];
  const float M3 = P[3], M4 = P[4], M5 = P[5];
  const float M6 = P[6], M7 = P[7], M8 = P[8];
  const float gamma = P[9], apply = P[10];
  const float ccx = P[11], ccy = P[12], cw2 = P[13], ch2 = P[14];
  const float wA0 = P[16 + lane], wA1 = P[48 + lane];

  // ---- projective warp ----
  const float xn = 2.f * (float)ix * invWm1 - 1.f;
  const float yn = 2.f * (float)iy * invHm1 - 1.f;
  const float sx = M0 * xn + M1 * yn + M2;
  const float sy = M3 * xn + M4 * yn + M5;
  const float sz = M6 * xn + M7 * yn + M8;
  const float rz = 1.f / sz;
  const float u = sx * rz, v = sy * rz;
  const float px = (u + 1.f) * 0.5f * (float)(W - 1);
  const float py = (v + 1.f) * 0.5f * (float)(H - 1);
  const float validf =
      (px >= 0.f && px <= (float)(W - 1) && py >= 0.f && py <= (float)(H - 1)) ? 1.f : 0.f;
  float x0f = fminf(fmaxf(floorf(px), 0.f), (float)(W - 1));
  float y0f = fminf(fmaxf(floorf(py), 0.f), (float)(H - 1));
  const float wx = px - x0f, wy = py - y0f;
  int x0 = min(max((int)x0f, 0), W - 1);
  int y0 = min(max((int)y0f, 0), H - 1);
  int x1 = min(x0 + 1, W - 1);
  int y1 = min(y0 + 1, H - 1);

  // ---- bilinear gather (L2-resident source) ----
  const float* base = img + (size_t)b * HW * 3;
  const float* p00 = base + ((size_t)y0 * W + x0) * 3;
  const float* p01 = base + ((size_t)y0 * W + x1) * 3;
  const float* p10 = base + ((size_t)y1 * W + x0) * 3;
  const float* p11 = base + ((size_t)y1 * W + x1) * 3;
  const float w00 = (1.f - wx) * (1.f - wy), w01 = wx * (1.f - wy);
  const float w10 = (1.f - wx) * wy,         w11 = wx * wy;
  float r  = (p00[0] * w00 + p01[0] * w01 + p10[0] * w10 + p11[0] * w11) * validf;
  float g  = (p00[1] * w00 + p01[1] * w01 + p10[1] * w10 + p11[1] * w11) * validf;
  float bl = (p00[2] * w00 + p01[2] * w01 + p10[2] * w10 + p11[2] * w11) * validf;

  // ---- gamma: clamp to [0,1] then exp2(g*log2(x)) on the trans pipe ----
  r  = pow01(fminf(fmaxf(r,  0.f), 1.f), gamma);
  g  = pow01(fminf(fmaxf(g,  0.f), 1.f), gamma);
  bl = pow01(fminf(fmaxf(bl, 0.f), 1.f), gamma);

  // ---- 3x3 color matrix on the matrix pipe ----
  float orr, og, ob;
#if __has_builtin(__builtin_amdgcn_wmma_f32_16x16x4_f32) && \
    __has_builtin(__builtin_amdgcn_ds_swizzle)
  {
    // B operand (4x16 f32): VGPR0 = row K=0 (lanes0-15) / K=2 (lanes16-31),
    // VGPR1 = K=1 / K=3.  Columns = pixels.  D: VGPRd lanes0-15 = channel d.
    const bool lo = (lane < 16);
    v2f Aop; Aop.x = wA0; Aop.y = wA1;
    v8f Z = {0.f, 0.f, 0.f, 0.f, 0.f, 0.f, 0.f, 0.f};
    // WMMA #1: pixels owned by lanes 0..15
    float bsw = swz_and15(bl);                       // lane L+16 <- b of lane L
    v2f B1; B1.x = lo ? r : bsw; B1.y = lo ? g : 0.f;
    v8f D1 = __builtin_amdgcn_wmma_f32_16x16x4_f32(
        false, Aop, false, B1, (short)0, Z, false, false);
    // WMMA #2: pixels owned by lanes 16..31 (swap halves into column slots)
    float rsw = swz_xor16(r);
    float gsw = swz_xor16(g);
    v2f B2; B2.x = lo ? rsw : bl; B2.y = lo ? gsw : 0.f;
    v8f D2 = __builtin_amdgcn_wmma_f32_16x16x4_f32(
        false, Aop, false, B2, (short)0, Z, false, false);
    // Results for upper-half pixels sit in lanes 0..15 of D2 -> move back.
    float r2 = swz_and15(D2[0]);
    float g2 = swz_and15(D2[1]);
    float b2 = swz_and15(D2[2]);
    orr = lo ? D1[0] : r2;
    og  = lo ? D1[1] : g2;
    ob  = lo ? D1[2] : b2;
  }
#else
  {
    orr = P[16 + 0] * r + P[48 + 0] * g + P[16 + 16] * bl;
    og  = P[16 + 1] * r + P[48 + 1] * g + P[16 + 17] * bl;
    ob  = P[16 + 2] * r + P[48 + 2] * g + P[16 + 18] * bl;
    (void)wA0; (void)wA1;
  }
#endif
  orr = fminf(fmaxf(orr, 0.f), 1.f);
  og  = fminf(fmaxf(og,  0.f), 1.f);
  ob  = fminf(fmaxf(ob,  0.f), 1.f);

  // ---- cutout (branch-free mask) ----
  const float gx = (float)ix * invWm1;
  const float gy = (float)iy * invHm1;
  const float cutm = (apply > 0.5f && fabsf(gx - ccx) < cw2 && fabsf(gy - ccy) < ch2)
                         ? 0.f : 1.f;
  orr *= cutm; og *= cutm; ob *= cutm;

  if (idx < HW) {
    float* o = out + (size_t)b * HW * 3 + (size_t)idx * 3;
    o[0] = orr; o[1] = og; o[2] = ob;
  }
}

extern "C" void kernel_launch(void* const* d_in, const int* in_sizes, int n_in,
                              void* d_out, int out_size, void* d_ws, size_t ws_size,
                              hipStream_t stream) {
  const float* images = (const float*)d_in[0];
  const float* geom   = (const float*)d_in[1];
  const float* color  = (const float*)d_in[2];
  const float* cut    = (const float*)d_in[3];
  float* out = (float*)d_out;
  float* ws  = (float*)d_ws;

  const int B  = in_sizes[1] / 8;                 // geom_u is (B,8)
  const int HW = in_sizes[0] / (B * 3);           // images is (B,H,W,3)
  int H = 512, W = 512;
  if (HW != 512 * 512) {
    int s = (int)(sqrtf((float)HW) + 0.5f);
    H = W = s;
  }
  int wshift = -1;
  if ((W & (W - 1)) == 0) { wshift = 0; while ((1 << wshift) < W) ++wshift; }

  aug_setup<<<(B + 31) / 32, 32, 0, stream>>>(geom, color, cut, ws, B);

  const int total = H * W;
  dim3 grid((total + 255) / 256, B);
  aug_main<<<grid, dim3(256), 0, stream>>>(
      images, ws, out, H, W, wshift,
      1.f / (float)(W - 1), 1.f / (float)(H - 1));
}